// KMeansBatch_70050916598123
// MI455X (gfx1250) — compile-verified
//
#include <hip/hip_runtime.h>

// Problem constants (match the JAX reference)
#define BATCH 8
#define NPTS  32768
#define DIM   64
#define KCL   128     // clusters
#define KM_ITERS 10

typedef float v2f __attribute__((ext_vector_type(2)));
typedef float v8f __attribute__((ext_vector_type(8)));

// ---------------------------------------------------------------------------
// Workspace layout (float units):
//   centers : [0,      65536)   B*K*D
//   c2      : [65536,  66560)   B*K       (cluster squared norms)
//   sums    : [66560,  132096)  B*K*D     (per-iter scatter sums)
//   counts  : [132096, 133120)  B*K
//   assign  : [133120, ...)     B*N int32
// Total ~1.55 MB.
// ---------------------------------------------------------------------------

// ---- init: deterministic strided pick of K points per batch + ||c||^2 ------
__global__ void km_init_centers(const float* __restrict__ x,
                                float* __restrict__ centers,
                                float* __restrict__ c2) {
    const int k = blockIdx.x, b = blockIdx.y, d = threadIdx.x;
    const int src = k * (NPTS / KCL);          // deterministic "random" seed pick
    const float v = x[((size_t)b * NPTS + src) * DIM + d];
    centers[((size_t)b * KCL + k) * DIM + d] = v;
    __shared__ float red[DIM];
    red[d] = v * v;
    __syncthreads();
    if (d == 0) {
        float s = 0.f;
        for (int i = 0; i < DIM; ++i) s += red[i];
        c2[b * KCL + k] = s;
    }
}

__global__ void km_zero(float* __restrict__ p, int n) {
    int i = blockIdx.x * blockDim.x + threadIdx.x;
    if (i < n) p[i] = 0.f;
}

// ---- fused assignment (WMMA f32 16x16x4 distance GEMM) + scatter-add -------
// Block = 256 threads (8 waves). Each wave owns 16 points; block owns 128
// points and keeps the full 128x64 center tile (32 KB) + c2 (512 B) in LDS.
__global__ __launch_bounds__(256)
void km_assign_scatter(const float* __restrict__ x,
                       const float* __restrict__ centers,
                       const float* __restrict__ c2g,
                       float* __restrict__ sums,
                       float* __restrict__ counts,
                       int* __restrict__ assign_out) {
    __shared__ float ldsC[KCL * DIM];   // 32 KB center tile for this batch
    __shared__ float ldsC2[KCL];

    const int b   = blockIdx.y;
    const int tid = threadIdx.x;

    // Cooperative load of centers into LDS (2048 float4 / 256 threads)
    const float4* cb4  = (const float4*)(centers + (size_t)b * KCL * DIM);
    float4*       lds4 = (float4*)ldsC;
    #pragma unroll
    for (int i = 0; i < (KCL * DIM / 4) / 256; ++i)
        lds4[tid + i * 256] = cb4[tid + i * 256];
    if (tid < KCL) ldsC2[tid] = c2g[b * KCL + tid];
    __syncthreads();

    const int wave   = tid >> 5;
    const int lane   = tid & 31;
    const int hi     = lane >> 4;      // which half of the wave
    const int lp     = lane & 15;
    const int base_n = (blockIdx.x * 8 + wave) * 16;

    // B-matrix fragments: 16 points x 64 dims resident in 32 VGPRs.
    // B is KxN (4x16): lanes 0-15 hold K={0,1} in v0/v1, lanes 16-31 K={2,3}.
    const float* xp = x + ((size_t)b * NPTS + base_n + lp) * DIM;
    v2f bfrag[16];
    #pragma unroll
    for (int kc = 0; kc < 16; ++kc)
        bfrag[kc] = *(const v2f*)(xp + kc * 4 + hi * 2);

    float best  = 3.4e38f;
    int   bestk = 0;

    // 8 cluster tiles x 16 chained K=4 steps = 128 WMMAs per wave
    #pragma unroll
    for (int kt = 0; kt < 8; ++kt) {
        v8f acc = {0.f, 0.f, 0.f, 0.f, 0.f, 0.f, 0.f, 0.f};
        const float* arow = ldsC + (kt * 16 + lp) * DIM + hi * 2;  // A: MxK, M=cluster
        #pragma unroll
        for (int kc = 0; kc < 16; ++kc) {
            v2f a = *(const v2f*)(arow + kc * 4);
            acc = __builtin_amdgcn_wmma_f32_16x16x4_f32(
                false, a, false, bfrag[kc], (short)0, acc, false, false);
        }
        // C/D layout: lane -> point (N), VGPR v (+8 for hi lanes) -> cluster (M)
        #pragma unroll
        for (int v = 0; v < 8; ++v) {
            const int kcl = kt * 16 + hi * 8 + v;
            const float s = ldsC2[kcl] - 2.0f * acc[v];   // ||c||^2 - 2 x.c
            if (s < best) { best = s; bestk = kcl; }
        }
    }

    // Merge the two cluster halves held by lane L and lane L^16 (same point)
    {
        const float ob = __shfl_xor(best, 16, 32);
        const int   ok = __shfl_xor(bestk, 16, 32);
        if (ob < best || (ob == best && ok < bestk)) { best = ob; bestk = ok; }
    }
    if (lane < 16)
        assign_out[(size_t)b * NPTS + base_n + lane] = bestk;

    // Scatter-add: whole wave cooperates per point (lane covers 2 dims)
    #pragma unroll 1
    for (int p = 0; p < 16; ++p) {
        const int k = __shfl(bestk, p, 32);
        const float* xq = x + ((size_t)b * NPTS + base_n + p) * DIM + lane * 2;
        const float v0 = xq[0], v1 = xq[1];
        float* s0 = sums + ((size_t)b * KCL + k) * DIM + lane * 2;
        atomicAdd(s0,     v0);
        atomicAdd(s0 + 1, v1);
        if (lane == 0) atomicAdd(&counts[b * KCL + k], 1.0f);
    }
}

// ---- center update: mean (or deterministic reseed if empty) + ||c||^2 ------
__global__ void km_update_centers(const float* __restrict__ x,
                                  const float* __restrict__ sums,
                                  const float* __restrict__ counts,
                                  float* __restrict__ centers,
                                  float* __restrict__ c2,
                                  int iter) {
    const int k = blockIdx.x, b = blockIdx.y, d = threadIdx.x;
    const float cnt = counts[b * KCL + k];
    float v;
    if (cnt > 0.0f) {
        v = sums[((size_t)b * KCL + k) * DIM + d] / cnt;
    } else {
        // deterministic reseed (stand-in for jax.random.randint)
        unsigned h = (unsigned)(b * 2654435761u) ^ (unsigned)(k * 40503u)
                   ^ (unsigned)(iter * 9973u);
        const int idx = (int)(h % (unsigned)NPTS);
        v = x[((size_t)b * NPTS + idx) * DIM + d];
    }
    centers[((size_t)b * KCL + k) * DIM + d] = v;
    __shared__ float red[DIM];
    red[d] = v * v;
    __syncthreads();
    if (d == 0) {
        float s = 0.f;
        for (int i = 0; i < DIM; ++i) s += red[i];
        c2[b * KCL + k] = s;
    }
}

// ---- pack outputs: [centers (B*K*D fp32) | assignments (B*N, as float)] ----
__global__ void km_write_out(const float* __restrict__ centers,
                             const int* __restrict__ assign_in,
                             float* __restrict__ out) {
    const int i = blockIdx.x * blockDim.x + threadIdx.x;
    if (i < BATCH * KCL * DIM) out[i] = centers[i];
    if (i < BATCH * NPTS)      out[BATCH * KCL * DIM + i] = (float)assign_in[i];
}

extern "C" void kernel_launch(void* const* d_in, const int* in_sizes, int n_in,
                              void* d_out, int out_size, void* d_ws, size_t ws_size,
                              hipStream_t stream) {
    (void)in_sizes; (void)n_in; (void)out_size; (void)ws_size;

    const float* x = (const float*)d_in[0];
    float* ws      = (float*)d_ws;
    float* centers = ws;                 // 65536 floats
    float* c2      = ws + 65536;         // 1024
    float* sums    = ws + 66560;         // 65536
    float* counts  = ws + 132096;        // 1024
    int*   assign  = (int*)(ws + 133120);// B*N int32
    float* out     = (float*)d_out;

    km_init_centers<<<dim3(KCL, BATCH), DIM, 0, stream>>>(x, centers, c2);

    for (int it = 0; it < KM_ITERS; ++it) {
        // sums + counts are contiguous: zero both in one launch (66560 floats)
        km_zero<<<(66560 + 255) / 256, 256, 0, stream>>>(sums, 66560);
        km_assign_scatter<<<dim3(NPTS / 128, BATCH), 256, 0, stream>>>(
            x, centers, c2, sums, counts, assign);
        km_update_centers<<<dim3(KCL, BATCH), DIM, 0, stream>>>(
            x, sums, counts, centers, c2, it + 1);
    }

    km_write_out<<<(BATCH * NPTS + 255) / 256, 256, 0, stream>>>(centers, assign, out);
}